// GAEExercise_88880053223613
// MI455X (gfx1250) — compile-verified
//
#include <hip/hip_runtime.h>
#include <hip/hip_bf16.h>

// GAE (generalized advantage estimation) as a blocked backward scan.
//   delta_t = r_t + GAMMA * v_{t+1} - v_t
//   A_t     = delta_t + c * A_{t+1},  c = GAMMA*LAM
//   ret_t   = A_t + v_t
// Block size 16 along time. Linearity: A(block) = localA(zero carry) + c^(16-j)*carry.
// Pass1: per-(row,block) zero-carry aggregate S. Pass2: tiny serial carry scan over
// 512 blocks per row. Pass3: 16x16 tiles via V_WMMA_F32_16X16X4_F32 (4 K-chunks +
// 1 carry-fold WMMA), all f32 so precision matches the reference.
// Outputs stored non-temporally so rewards/values stay resident in the 192MB L2
// between pass 1 and pass 3.

#define BATCH 2048
#define SEQL  8192
#define NB    512            // SEQL / 16
#define GAM   0.99f

typedef __attribute__((ext_vector_type(2))) float v2f;
typedef __attribute__((ext_vector_type(8))) float v8f;

constexpr double dpow(double x, int n) { double p = 1.0; for (int i = 0; i < n; ++i) p *= x; return p; }
constexpr double CD   = 0.99 * 0.95;         // gamma * lam in double, rounded once
constexpr float  CC   = (float)CD;
constexpr float  CC2  = (float)dpow(CD, 2);
constexpr float  CC4  = (float)dpow(CD, 4);
constexpr float  CC8  = (float)dpow(CD, 8);
constexpr float  CC16 = (float)dpow(CD, 16);

// Branch-free: c^e for e >= 0, 0.0f for e < 0. Each factor is a two-constant
// select -> guaranteed v_cndmask + v_mul, no EXEC divergence on the WMMA path.
__device__ __forceinline__ float powc_gated(int e) {
  const int   em   = e & 31;
  const float f1   = (em & 1)  ? CC   : 1.f;
  const float f2   = (em & 2)  ? CC2  : 1.f;
  const float f4   = (em & 4)  ? CC4  : 1.f;
  const float f8   = (em & 8)  ? CC8  : 1.f;
  const float f16  = (em & 16) ? CC16 : 1.f;
  const float gate = (e >= 0)  ? 1.f  : 0.f;
  return ((gate * f1) * (f2 * f4)) * (f8 * f16);
}

// ---------------------------------------------------------------------------
// Pass 1: S_t[b][r] = zero-carry scan value at the start of 16-block b of row r.
// Thread g -> (r = g & 2047, b = g >> 11). Lanes consecutive in r: each lane
// streams its own 64B cacheline (fully consumed); S_t write is coalesced.
// ---------------------------------------------------------------------------
__global__ __launch_bounds__(256) void gae_block_agg(
    const float* __restrict__ rw, const float* __restrict__ vl,
    const float* __restrict__ nv, float* __restrict__ S_t) {
  const int g = blockIdx.x * 256 + threadIdx.x;
  const int r = g & (BATCH - 1);
  const int b = g >> 11;
  const float* rp = rw + (size_t)r * SEQL + b * 16;
  const float* vp = vl + (size_t)r * SEQL + b * 16;
  float rv[16], vv[17];
  const float4* rp4 = (const float4*)rp;
  const float4* vp4 = (const float4*)vp;
#pragma unroll
  for (int q = 0; q < 4; ++q) {
    float4 t = rp4[q];
    rv[4*q] = t.x; rv[4*q+1] = t.y; rv[4*q+2] = t.z; rv[4*q+3] = t.w;
    float4 u = vp4[q];
    vv[4*q] = u.x; vv[4*q+1] = u.y; vv[4*q+2] = u.z; vv[4*q+3] = u.w;
  }
  vv[16] = (b == NB - 1) ? nv[r] : vp[16];
  // Horner from the right with zero carry (same op order as the reference scan).
  float s = 0.f;
#pragma unroll
  for (int k = 15; k >= 0; --k)
    s = (rv[k] + GAM * vv[k + 1] - vv[k]) + CC * s;
  S_t[(size_t)b * BATCH + r] = s;
}

// ---------------------------------------------------------------------------
// Pass 2: carry into each block: C_b = S_{b+1} + c^16 * C_{b+1}, C_{NB-1} = 0.
// One lane per row; loads/stores coalesced across lanes at each b.
// ---------------------------------------------------------------------------
__global__ __launch_bounds__(256) void gae_carry_scan(
    const float* __restrict__ S_t, float* __restrict__ C_t) {
  const int r = blockIdx.x * 256 + threadIdx.x;
  float c = 0.f;
#pragma unroll 8
  for (int b = NB - 1; b >= 0; --b) {
    C_t[(size_t)b * BATCH + r] = c;
    c = S_t[(size_t)b * BATCH + r] + CC16 * c;
  }
}

// ---------------------------------------------------------------------------
// Pass 3: final tiles. WG = 16 rows x 128 timesteps (8 waves, one 16x16 chunk
// per wave). values staged in LDS (stride 132 floats: 16B-aligned rows,
// conflict-free fragment gather), deltas computed into LDS, then
// D[r][j] = sum_k Delta[r][k] * W[k][j] + c^(16-j)*carry via 5 f32 WMMAs.
// ---------------------------------------------------------------------------
__global__ __launch_bounds__(256) void gae_wmma_out(
    const float* __restrict__ rw, const float* __restrict__ vl,
    const float* __restrict__ nv, const float* __restrict__ C_t,
    float* __restrict__ adv, float* __restrict__ ret) {
  __shared__ float v_s[16][132];
  __shared__ float d_s[16][132];

  const int tid  = threadIdx.x;
  const int st   = blockIdx.x & 63;   // supertile index along time (128 cols each)
  const int rb   = blockIdx.x >> 6;   // 16-row block
  const int r0   = rb * 16;
  const int tbse = st * 128;

  // Cooperative coalesced load of the values tile (cols 0..127).
#pragma unroll
  for (int s = 0; s < 2; ++s) {
    const int f   = tid + 256 * s;
    const int row = f >> 5;
    const int c4  = (f & 31) * 4;
    const float4 u = *(const float4*)(vl + (size_t)(r0 + row) * SEQL + tbse + c4);
    *(float4*)&v_s[row][c4] = u;
  }
  if (tid < 16) {
    v_s[tid][128] = (tbse + 128 == SEQL)
                        ? nv[r0 + tid]
                        : vl[(size_t)(r0 + tid) * SEQL + tbse + 128];
  }
  __syncthreads();

  // delta tile into LDS (rewards streamed straight from global, coalesced).
#pragma unroll
  for (int s = 0; s < 2; ++s) {
    const int f   = tid + 256 * s;
    const int row = f >> 5;
    const int c4  = (f & 31) * 4;
    const float4 rr = *(const float4*)(rw + (size_t)(r0 + row) * SEQL + tbse + c4);
    const float v0 = v_s[row][c4 + 0], v1 = v_s[row][c4 + 1];
    const float v2 = v_s[row][c4 + 2], v3 = v_s[row][c4 + 3];
    const float v4 = v_s[row][c4 + 4];
    float4 d;
    d.x = rr.x + GAM * v1 - v0;
    d.y = rr.y + GAM * v2 - v1;
    d.z = rr.z + GAM * v3 - v2;
    d.w = rr.w + GAM * v4 - v3;
    *(float4*)&d_s[row][c4] = d;
  }
  __syncthreads();

  const int lane = tid & 31;
  const int w    = tid >> 5;       // wave index = 16-chunk within supertile
  const int j    = lane & 15;      // column (time-within-chunk) / A-row index
  const int half = lane >> 4;
  const int tloc = w * 16;
  const int tb   = st * 8 + w;     // global 16-block index

  // Branch-free carry fetch keeps EXEC all-ones for WMMA.
  const float cl    = C_t[(size_t)tb * BATCH + r0 + j];
  const float carry = half ? 0.f : cl;

  v8f acc = {0.f, 0.f, 0.f, 0.f, 0.f, 0.f, 0.f, 0.f};
#pragma unroll
  for (int kk = 0; kk < 4; ++kk) {
    const int k0 = kk * 4 + half * 2;          // A/B K-split per lane-half
    v2f a;                                      // A = Delta, M=j(row), K=k0..k0+1
    a.x = d_s[j][tloc + k0];
    a.y = d_s[j][tloc + k0 + 1];
    v2f bf;                                     // B = W[k][j] = c^(k-j), k>=j
    bf.x = powc_gated(k0 - j);
    bf.y = powc_gated(k0 + 1 - j);
    acc = __builtin_amdgcn_wmma_f32_16x16x4_f32(
        false, a, false, bf, (short)0, acc, false, false);
  }
  {
    // Carry fold: rank-1 update, A col0 = carry per row, B row0 = c^(16-j).
    v2f a;  a.x  = carry;  a.y  = 0.f;
    v2f bf;
    bf.x = (half ? 0.f : 1.f) * powc_gated(16 - j);
    bf.y = 0.f;
    acc = __builtin_amdgcn_wmma_f32_16x16x4_f32(
        false, a, false, bf, (short)0, acc, false, false);
  }

  // D layout: VGPR v -> rows (v, v+8) split across lane halves, col = j.
  // Non-temporal stores: outputs are write-once; keep rewards/values in L2.
#pragma unroll
  for (int v = 0; v < 8; ++v) {
    const int row    = v + half * 8;
    const size_t idx = (size_t)(r0 + row) * SEQL + tbse + tloc + j;
    const float a    = acc[v];
    __builtin_nontemporal_store(a, adv + idx);
    __builtin_nontemporal_store(a + v_s[row][tloc + j], ret + idx);
  }
}

// ---------------------------------------------------------------------------
extern "C" void kernel_launch(void* const* d_in, const int* in_sizes, int n_in,
                              void* d_out, int out_size, void* d_ws, size_t ws_size,
                              hipStream_t stream) {
  (void)in_sizes; (void)n_in; (void)out_size; (void)ws_size;
  const float* rw = (const float*)d_in[0];   // rewards     (B, S)
  const float* vl = (const float*)d_in[1];   // values      (B, S)
  const float* nv = (const float*)d_in[2];   // next_values (B,)

  float* adv = (float*)d_out;                          // advantages (B, S)
  float* ret = adv + (size_t)BATCH * SEQL;             // returns    (B, S)

  float* S_t = (float*)d_ws;                           // [NB][BATCH]  4 MB
  float* C_t = S_t + (size_t)NB * BATCH;               // [NB][BATCH]  4 MB

  gae_block_agg<<<(BATCH * NB) / 256, 256, 0, stream>>>(rw, vl, nv, S_t);
  gae_carry_scan<<<BATCH / 256, 256, 0, stream>>>(S_t, C_t);
  gae_wmma_out<<<(BATCH / 16) * (SEQL / 128), 256, 0, stream>>>(
      rw, vl, nv, C_t, adv, ret);
}